// MambaBlock_2731599200629
// MI455X (gfx1250) — compile-verified
//
#include <hip/hip_runtime.h>

typedef __attribute__((ext_vector_type(2))) float v2f;
typedef __attribute__((ext_vector_type(8))) float v8f;

#define D_MODEL 128
#define D_INNER 256
#define D_STATE 16
#define DT_RANK 8
#define T_LEN   64
#define EPS_LN  1e-5f

// LDS row strides (padded for bank-conflict-free WMMA A-fragment b64 loads)
#define XS 132   // x_in / u: [64][132]
#define IS 260   // xact (xc -> act -> y): [64][260]
#define DSS 48   // dbl (dt|B|C): [64][48], cols 0..39 valid
#define OS 66    // output staging in s_u reuse: [128][66]

#define WMMA4(a, b, c) \
  __builtin_amdgcn_wmma_f32_16x16x4_f32(false, (a), false, (b), (short)0, (c), false, false)

// Generic pointer to LDS raw offset: generic LDS ptr = (shared_aperture_hi32 | offset32)
__device__ __forceinline__ unsigned lds_off(const void* p) {
  return (unsigned)(unsigned long long)p;
}

// CDNA5 async DMA: memory -> LDS, 4B per lane, per-lane LDS dest address (transpose-capable)
__device__ __forceinline__ void async_ld_b32(unsigned ldsoff, const float* g) {
  asm volatile("global_load_async_to_lds_b32 %0, %1, off th:TH_LOAD_NT"
               :: "v"(ldsoff), "v"(g) : "memory");
}
// CDNA5 async DMA: LDS -> memory, 8B per lane
__device__ __forceinline__ void async_st_b64(float* g, unsigned ldsoff) {
  asm volatile("global_store_async_from_lds_b64 %0, %1, off th:TH_STORE_NT"
               :: "v"(g), "v"(ldsoff) : "memory");
}
__device__ __forceinline__ void wait_async0() {
  asm volatile("s_wait_asynccnt 0x0" ::: "memory");
}

__global__ __launch_bounds__(256)
void mamba_block_kernel(const float* __restrict__ x,       // [4096][128][64]
                        const float* __restrict__ gamma,   // [128]
                        const float* __restrict__ beta,    // [128]
                        const float* __restrict__ W_in,    // [128][512]
                        const float* __restrict__ conv_w,  // [256][4]
                        const float* __restrict__ conv_b,  // [256]
                        const float* __restrict__ W_x,     // [256][40]
                        const float* __restrict__ W_dt,    // [8][256]
                        const float* __restrict__ b_dt,    // [256]
                        const float* __restrict__ A_log,   // [256][16]
                        const float* __restrict__ Dskip,   // [256]
                        const float* __restrict__ W_out,   // [256][128]
                        float* __restrict__ out)           // [4096][128][64]
{
  __shared__ float s_xin[T_LEN * XS];   // 8448 f : x tile [t][f] (kept for residual)
  __shared__ float s_u[T_LEN * XS];     // 8448 f : LN output; reused as out stage [f][t] (128*66=8448)
  __shared__ float s_xact[T_LEN * IS];  // 16640 f: xc -> conv/silu -> y
  __shared__ float s_dbl[T_LEN * DSS];  // 3072 f : dt(8) | B(16) | C(16)
  __shared__ float s_red[512];
  __shared__ float s_mu[T_LEN];
  __shared__ float s_rs[T_LEN];

  const int tid  = threadIdx.x;
  const int lane = tid & 31;
  const int wv   = tid >> 5;     // wave id 0..7
  const int ln   = lane & 15;    // N / M index within fragment
  const int hi   = lane >> 4;    // half-wave select
  const int koff = 2 * hi;       // K offset for upper half lanes

  const long seq = blockIdx.x;
  const float* xg = x + seq * (long)(D_MODEL * T_LEN);
  float* og = out + seq * (long)(D_MODEL * T_LEN);

  // ------- async DMA load of x tile, transposed [f][t] -> [t][f] (ASYNCcnt path) -------
  #pragma unroll
  for (int r = 0; r < 32; ++r) {
    int e = r * 256 + tid;             // coalesced over f*64+t (128B/lane-group)
    int f = e >> 6, t = e & 63;
    async_ld_b32(lds_off(&s_xin[t * XS + f]), xg + e);
  }
  wait_async0();
  __syncthreads();

  // ---------------- layernorm over f (128) per t ----------------
  {
    int t = tid & 63, q = tid >> 6;    // 4 partial sums per row
    float s = 0.f, s2 = 0.f;
    const float* row = &s_xin[t * XS + q * 32];
    #pragma unroll
    for (int f = 0; f < 32; ++f) { float v = row[f]; s += v; s2 += v * v; }
    s_red[tid] = s;
    s_red[256 + tid] = s2;
  }
  __syncthreads();
  if (tid < 64) {
    float s = 0.f, s2 = 0.f;
    #pragma unroll
    for (int q = 0; q < 4; ++q) { s += s_red[q * 64 + tid]; s2 += s_red[256 + q * 64 + tid]; }
    float mu = s * (1.f / 128.f);
    float var = s2 * (1.f / 128.f) - mu * mu;
    s_mu[tid] = mu;
    s_rs[tid] = rsqrtf(var + EPS_LN);
  }
  __syncthreads();
  for (int r = 0; r < 32; ++r) {
    int e = r * 256 + tid;
    int t = e >> 7, f = e & 127;
    float v = s_xin[t * XS + f];
    s_u[t * XS + f] = (v - s_mu[t]) * s_rs[t] * gamma[f] + beta[f];
  }
  __syncthreads();

  // ---------------- GEMM1: u(64x128) @ W_in[:, :256] -> xc ----------------
  for (int nt = wv; nt < 16; nt += 8) {
    int n0 = nt * 16;
    v8f a0c = {}, a1c = {}, a2c = {}, a3c = {};
    for (int k0 = 0; k0 < 128; k0 += 4) {
      int kk = k0 + koff;
      v2f b; b.x = W_in[kk * 512 + n0 + ln]; b.y = W_in[(kk + 1) * 512 + n0 + ln];
      v2f a0 = *(const v2f*)&s_u[(0 * 16 + ln) * XS + kk];
      v2f a1 = *(const v2f*)&s_u[(1 * 16 + ln) * XS + kk];
      v2f a2 = *(const v2f*)&s_u[(2 * 16 + ln) * XS + kk];
      v2f a3 = *(const v2f*)&s_u[(3 * 16 + ln) * XS + kk];
      a0c = WMMA4(a0, b, a0c); a1c = WMMA4(a1, b, a1c);
      a2c = WMMA4(a2, b, a2c); a3c = WMMA4(a3, b, a3c);
    }
    int c0 = n0 + ln;
    v8f accs[4] = {a0c, a1c, a2c, a3c};
    #pragma unroll
    for (int m = 0; m < 4; ++m)
      #pragma unroll
      for (int r = 0; r < 8; ++r)
        s_xact[(m * 16 + r + 8 * hi) * IS + c0] = accs[m][r];
  }
  __syncthreads();

  // ---------------- depthwise causal conv (width 4) + SiLU, in place ----------------
  {
    int c = tid;
    float cw0 = conv_w[c * 4 + 0], cw1 = conv_w[c * 4 + 1];
    float cw2 = conv_w[c * 4 + 2], cw3 = conv_w[c * 4 + 3];
    float cb = conv_b[c];
    float w0 = 0.f, w1 = 0.f, w2 = 0.f;
    for (int t = 0; t < T_LEN; ++t) {
      float xt = s_xact[t * IS + c];
      float cv = w0 * cw0 + w1 * cw1 + w2 * cw2 + xt * cw3 + cb;
      s_xact[t * IS + c] = cv / (1.f + __expf(-cv));   // silu
      w0 = w1; w1 = w2; w2 = xt;
    }
  }
  __syncthreads();

  // ---------------- GEMM2: xact(64x256) @ W_x(256x40) -> dbl (N padded to 48) ----------------
  for (int tile = wv; tile < 12; tile += 8) {
    int mt = tile & 3, nt = tile >> 2;
    int col = nt * 16 + ln;
    bool okc = (col < DT_RANK + 2 * D_STATE);
    v8f acc = {};
    for (int k0 = 0; k0 < 256; k0 += 4) {
      int kk = k0 + koff;
      v2f a = *(const v2f*)&s_xact[(mt * 16 + ln) * IS + kk];
      v2f b;
      b.x = okc ? W_x[kk * 40 + col] : 0.f;
      b.y = okc ? W_x[(kk + 1) * 40 + col] : 0.f;
      acc = WMMA4(a, b, acc);
    }
    #pragma unroll
    for (int r = 0; r < 8; ++r)
      s_dbl[(mt * 16 + r + 8 * hi) * DSS + col] = acc[r];
  }

  // ---------------- per-thread scan setup (channel i = tid) ----------------
  float Wd[DT_RANK];
  #pragma unroll
  for (int r = 0; r < DT_RANK; ++r) Wd[r] = W_dt[r * 256 + tid];
  float bdt = b_dt[tid];
  float An[D_STATE];
  #pragma unroll
  for (int s = 0; s < D_STATE; ++s) An[s] = -__expf(A_log[tid * 16 + s]);
  float dsk = Dskip[tid];
  float h[D_STATE];
  #pragma unroll
  for (int s = 0; s < D_STATE; ++s) h[s] = 0.f;
  __syncthreads();   // dbl complete; GEMM2 done reading xact

  // ---------------- selective scan: no barriers (channels independent) ----------------
  for (int t = 0; t < T_LEN; ++t) {
    const float* drow = &s_dbl[t * DSS];
    float dtv = bdt;
    #pragma unroll
    for (int r = 0; r < DT_RANK; ++r) dtv += drow[r] * Wd[r];
    float d = (dtv > 20.f) ? dtv : log1pf(__expf(dtv));   // softplus
    float xt = s_xact[t * IS + tid];
    float dx = d * xt;
    float acc = 0.f;
    #pragma unroll
    for (int s = 0; s < D_STATE; ++s) {
      float dA = __expf(d * An[s]);
      h[s] = dA * h[s] + dx * drow[8 + s];
      acc += h[s] * drow[24 + s];
    }
    s_xact[t * IS + tid] = acc + xt * dsk;   // y (gated below)
  }
  __syncthreads();

  // ---------------- z-GEMM replay: u @ W_in[:, 256:], gate y *= silu(z) ----------------
  for (int nt = wv; nt < 16; nt += 8) {
    int n0 = nt * 16;
    v8f a0c = {}, a1c = {}, a2c = {}, a3c = {};
    for (int k0 = 0; k0 < 128; k0 += 4) {
      int kk = k0 + koff;
      v2f b; b.x = W_in[kk * 512 + 256 + n0 + ln]; b.y = W_in[(kk + 1) * 512 + 256 + n0 + ln];
      v2f a0 = *(const v2f*)&s_u[(0 * 16 + ln) * XS + kk];
      v2f a1 = *(const v2f*)&s_u[(1 * 16 + ln) * XS + kk];
      v2f a2 = *(const v2f*)&s_u[(2 * 16 + ln) * XS + kk];
      v2f a3 = *(const v2f*)&s_u[(3 * 16 + ln) * XS + kk];
      a0c = WMMA4(a0, b, a0c); a1c = WMMA4(a1, b, a1c);
      a2c = WMMA4(a2, b, a2c); a3c = WMMA4(a3, b, a3c);
    }
    int c0 = n0 + ln;
    v8f accs[4] = {a0c, a1c, a2c, a3c};
    #pragma unroll
    for (int m = 0; m < 4; ++m)
      #pragma unroll
      for (int r = 0; r < 8; ++r) {
        int tq = m * 16 + r + 8 * hi;
        float zv = accs[m][r];
        s_xact[tq * IS + c0] *= zv / (1.f + __expf(-zv));   // y *= silu(z)
      }
  }
  __syncthreads();

  // ------- GEMM3: y(64x256) @ W_out(256x128) -> stage [f][t] in s_u, residual folded -------
  {
    int n0 = wv * 16;
    v8f a0c = {}, a1c = {}, a2c = {}, a3c = {};
    for (int k0 = 0; k0 < 256; k0 += 4) {
      int kk = k0 + koff;
      v2f b; b.x = W_out[kk * 128 + n0 + ln]; b.y = W_out[(kk + 1) * 128 + n0 + ln];
      v2f a0 = *(const v2f*)&s_xact[(0 * 16 + ln) * IS + kk];
      v2f a1 = *(const v2f*)&s_xact[(1 * 16 + ln) * IS + kk];
      v2f a2 = *(const v2f*)&s_xact[(2 * 16 + ln) * IS + kk];
      v2f a3 = *(const v2f*)&s_xact[(3 * 16 + ln) * IS + kk];
      a0c = WMMA4(a0, b, a0c); a1c = WMMA4(a1, b, a1c);
      a2c = WMMA4(a2, b, a2c); a3c = WMMA4(a3, b, a3c);
    }
    __syncthreads();   // everyone done reading s_u (z-GEMM) before overwrite
    int f0 = n0 + ln;
    v8f accs[4] = {a0c, a1c, a2c, a3c};
    #pragma unroll
    for (int m = 0; m < 4; ++m)
      #pragma unroll
      for (int r = 0; r < 8; ++r) {
        int tq = m * 16 + r + 8 * hi;
        s_u[f0 * OS + tq] = accs[m][r] + s_xin[tq * XS + f0];   // + residual x_in
      }
  }
  __syncthreads();

  // ------- async DMA store: LDS [f][t] -> global [f][t], 8B/lane (ASYNCcnt path) -------
  #pragma unroll
  for (int r = 0; r < 16; ++r) {
    int e = (r * 256 + tid) * 2;       // even element index: f = e/64, t = e%64 (t even)
    int f = e >> 6, t = e & 63;
    async_st_b64(og + e, lds_off(&s_u[f * OS + t]));
  }
  wait_async0();   // (S_ENDPGM also implies wait-idle)
}

extern "C" void kernel_launch(void* const* d_in, const int* in_sizes, int n_in,
                              void* d_out, int out_size, void* d_ws, size_t ws_size,
                              hipStream_t stream) {
  const float* x      = (const float*)d_in[0];
  const float* gamma  = (const float*)d_in[1];
  const float* beta   = (const float*)d_in[2];
  const float* W_in   = (const float*)d_in[3];
  const float* conv_w = (const float*)d_in[4];
  const float* conv_b = (const float*)d_in[5];
  const float* W_x    = (const float*)d_in[6];
  const float* W_dt   = (const float*)d_in[7];
  const float* b_dt   = (const float*)d_in[8];
  const float* A_log  = (const float*)d_in[9];
  const float* Dskip  = (const float*)d_in[10];
  const float* W_out  = (const float*)d_in[11];
  float* out = (float*)d_out;

  dim3 grid(16 * 256);   // one workgroup per sequence
  dim3 block(256);       // 8 wave32
  mamba_block_kernel<<<grid, block, 0, stream>>>(x, gamma, beta, W_in, conv_w, conv_b,
                                                 W_x, W_dt, b_dt, A_log, Dskip, W_out, out);
}